// StableSSMMemory_42279658062192
// MI455X (gfx1250) — compile-verified
//
#include <hip/hip_runtime.h>
#include <cmath>

// ---- problem constants (reference: T,N=8192,4096; H,P,R=64,6,4) ----
#define T_STEPS 8192
#define N_DIM   4096
#define H_DIM   64
#define P_DIM   6
#define SEG     16                 // scan chunk length
#define NSEG    (T_STEPS / SEG)    // 512 segments
#define SEG_BLOCKS (NSEG / 16)     // 32 blocks, 16 segment-rows each

typedef __attribute__((ext_vector_type(2))) float v2f;
typedef __attribute__((ext_vector_type(4))) float v4f;
typedef __attribute__((ext_vector_type(8))) float v8f;

// V_WMMA_F32_16X16X4_F32: D(16x16,f32) = A(16x4,f32) x B(4x16,f32) + C
__device__ __forceinline__ v8f wmma4(v2f a, v2f b, v8f c) {
  return __builtin_amdgcn_wmma_f32_16x16x4_f32(false, a, false, b, (short)0, c,
                                               false, false);
}

__device__ __forceinline__ float wave_reduce(float v) {
  #pragma unroll
  for (int off = 16; off > 0; off >>= 1) v += __shfl_down(v, off, 32);
  return v;
}

// ---------------------------------------------------------------------------
// K1: summarise one time-row (reduce over N=4096) + fused Bu = dt * u @ B^T.
// Carries ~403 MB of the ~406 MB total traffic (streamed once; > 192 MB L2)
// -> non-temporal b128 loads so the stream doesn't thrash L2, which we want
// to keep for Bu / h_hist reuse in later phases. HBM floor ~17 us @ 23.3 TB/s.
// ---------------------------------------------------------------------------
__global__ void summarize_kernel(const float* __restrict__ omega,
                                 const float* __restrict__ phi,
                                 const float* __restrict__ amp,
                                 const float* __restrict__ Bmat,   // [H,P]
                                 const float* __restrict__ dtp,
                                 float* __restrict__ u_out,        // [T,6] (d_out)
                                 float* __restrict__ u_ws,         // [T,8] (ws)
                                 float* __restrict__ Bu) {         // [T,64] (ws)
  const int t   = blockIdx.x;
  const int tid = threadIdx.x;                      // 256 threads = 8 waves
  const float* om = omega + (size_t)t * N_DIM;
  const float* ph = phi   + (size_t)t * N_DIM;
  const float* am = amp   + (size_t)t * N_DIM;

  float s_amp = 0.f, s_cos = 0.f, s_sin = 0.f, s_om = 0.f;
  float s_phi = 0.f, s_phi2 = 0.f, s_amp2 = 0.f;

  // 4096 elems / 256 threads / float4 = 4 iterations of NT b128 loads
  for (int it = 0; it < N_DIM / (256 * 4); ++it) {
    const int base = (it * 256 + tid) * 4;
    v4f o4 = __builtin_nontemporal_load((const v4f*)(om + base));
    v4f p4 = __builtin_nontemporal_load((const v4f*)(ph + base));
    v4f a4 = __builtin_nontemporal_load((const v4f*)(am + base));
    #pragma unroll
    for (int k = 0; k < 4; ++k) {
      float ov = o4[k], pv = p4[k], av = a4[k];
      float s, c;
      __sincosf(pv, &s, &c);
      s_amp  += av;
      s_cos  += c;
      s_sin  += s;
      s_om   += ov;
      s_phi  += pv;
      s_phi2 += pv * pv;
      s_amp2 += av * av;
    }
  }

  __shared__ float part[7][8];
  const int wid = tid >> 5, lid = tid & 31;
  float vals[7] = {s_amp, s_cos, s_sin, s_om, s_phi, s_phi2, s_amp2};
  #pragma unroll
  for (int q = 0; q < 7; ++q) {
    float r = wave_reduce(vals[q]);
    if (lid == 0) part[q][wid] = r;
  }
  __syncthreads();
  float tot[7];
  #pragma unroll
  for (int q = 0; q < 7; ++q) {
    float s = 0.f;
    #pragma unroll
    for (int w = 0; w < 8; ++w) s += part[q][w];
    tot[q] = s;
  }

  const float inv = 1.0f / (float)N_DIM;
  float u[6];
  u[0] = tot[0] * inv;                       // amp mean
  u[1] = tot[1] * inv;                       // cos(phi) mean
  u[2] = tot[2] * inv;                       // sin(phi) mean
  u[3] = tot[3] * inv;                       // omega mean
  float mphi = tot[4] * inv;
  u[4] = sqrtf(fmaxf(tot[5] * inv - mphi * mphi, 0.0f));  // population std
  u[5] = 0.5f * tot[6];                      // 0.5*sum(amp^2)

  if (tid < 6) {
    u_out[t * 6 + tid] = u[tid];
    u_ws[t * 8 + tid]  = u[tid];
  }
  if (tid < H_DIM) {                         // Bu[t][h] = dt * sum_p u[p]*B[h][p]
    const float dt = dtp[0];
    float acc = 0.f;
    #pragma unroll
    for (int p = 0; p < P_DIM; ++p) acc += u[p] * Bmat[tid * P_DIM + p];
    Bu[t * H_DIM + tid] = dt * acc;
  }
}

// ---------------------------------------------------------------------------
// K2: W = M^T where M = I + dt*(diag(-softplus(lambda)) + U V^T);
//     Cw = padded C^T [64 x 16] (cols >=6 zero).
// ---------------------------------------------------------------------------
__global__ void build_kernel(const float* __restrict__ lam,
                             const float* __restrict__ U,
                             const float* __restrict__ V,
                             const float* __restrict__ C,   // [P,H]
                             const float* __restrict__ dtp,
                             float* __restrict__ W,         // [64,64] = M^T
                             float* __restrict__ Cw) {      // [64,16]
  const int idx = blockIdx.x * 256 + threadIdx.x;
  const float dt = dtp[0];
  if (idx < 64 * 64) {
    const int i = idx >> 6, j = idx & 63;
    float a = 0.f;
    #pragma unroll
    for (int r = 0; r < 4; ++r) a += U[i * 4 + r] * V[j * 4 + r];
    float x  = lam[i];
    float sp = (x > 20.0f) ? x : log1pf(expf(x));   // softplus
    float m  = dt * a + ((i == j) ? (1.0f - dt * sp) : 0.0f);
    W[j * 64 + i] = m;                              // transpose
  }
  const int c = idx - 64 * 64;
  if (c >= 0 && c < 64 * 16) {
    const int k = c >> 4, n = c & 15;
    Cw[k * 16 + n] = (n < P_DIM) ? C[n * H_DIM + k] : 0.0f;
  }
}

// ---------------------------------------------------------------------------
// K3: Out = A @ B, 64x64x64 f32 via WMMA. 16 waves, one 16x16 tile each.
// Used to square W: W^2, W^4, W^8, W^16.
// ---------------------------------------------------------------------------
__global__ void gemm64_kernel(const float* __restrict__ A,
                              const float* __restrict__ B,
                              float* __restrict__ Out) {
  const int tid  = threadIdx.x;            // 512 = 16 waves
  const int wave = tid >> 5, lane = tid & 31;
  const int mt = wave >> 2, nt = wave & 3;
  const int half = lane >> 4, l15 = lane & 15;
  v8f acc = {0.f, 0.f, 0.f, 0.f, 0.f, 0.f, 0.f, 0.f};
  #pragma unroll
  for (int kt = 0; kt < 16; ++kt) {
    v2f a, b;
    const int ar = mt * 16 + l15;
    a.x = A[ar * 64 + kt * 4 + half * 2 + 0];
    a.y = A[ar * 64 + kt * 4 + half * 2 + 1];
    const int bn = nt * 16 + l15;
    b.x = B[(kt * 4 + half * 2 + 0) * 64 + bn];
    b.y = B[(kt * 4 + half * 2 + 1) * 64 + bn];
    acc = wmma4(a, b, acc);
  }
  #pragma unroll
  for (int r = 0; r < 8; ++r)
    Out[(mt * 16 + half * 8 + r) * 64 + nt * 16 + l15] = acc[r];
}

// ---------------------------------------------------------------------------
// K4: per-segment v: v <- v@W + bu_t for t = seg*16 .. seg*16+15.
// Block owns 16 segment-rows (independent across blocks); 4 waves cover the
// 64 state columns; 15 batched [16x64]@[64x64] WMMA steps.
// ---------------------------------------------------------------------------
__global__ void segment_v_kernel(const float* __restrict__ Bu,
                                 const float* __restrict__ W,
                                 float* __restrict__ Vseg) {
  __shared__ float Wl[64 * 64];
  __shared__ float Vt[16 * 65];            // stride 65: conflict-free A reads
  const int tid = threadIdx.x;             // 128 = 4 waves
  for (int i = tid; i < 64 * 64; i += 128) Wl[i] = W[i];
  const int seg0 = blockIdx.x * 16;
  for (int i = tid; i < 16 * 64; i += 128) {
    const int r = i >> 6, c = i & 63;
    Vt[r * 65 + c] = Bu[((seg0 + r) * SEG + 0) * 64 + c];   // v after step 0
  }
  __syncthreads();
  const int wave = tid >> 5, lane = tid & 31;
  const int nt = wave, half = lane >> 4, l15 = lane & 15;
  for (int j = 1; j < SEG; ++j) {
    v8f acc;
    #pragma unroll
    for (int r = 0; r < 8; ++r)            // C := bu_{t=seg*16+j}
      acc[r] = Bu[((seg0 + half * 8 + r) * SEG + j) * 64 + nt * 16 + l15];
    #pragma unroll
    for (int kt = 0; kt < 16; ++kt) {
      v2f a, b;
      a.x = Vt[l15 * 65 + kt * 4 + half * 2 + 0];
      a.y = Vt[l15 * 65 + kt * 4 + half * 2 + 1];
      b.x = Wl[(kt * 4 + half * 2 + 0) * 64 + nt * 16 + l15];
      b.y = Wl[(kt * 4 + half * 2 + 1) * 64 + nt * 16 + l15];
      acc = wmma4(a, b, acc);
    }
    __syncthreads();
    #pragma unroll
    for (int r = 0; r < 8; ++r)
      Vt[(half * 8 + r) * 65 + nt * 16 + l15] = acc[r];
    __syncthreads();
  }
  for (int i = tid; i < 16 * 64; i += 128) {
    const int r = i >> 6, c = i & 63;
    Vseg[(seg0 + r) * 64 + c] = Vt[r * 65 + c];
  }
}

// ---------------------------------------------------------------------------
// K5: sequential combine over 512 segments: h <- h@W16 + v_seg.
// Also records Hstart[seg] = h before segment seg. Depth 512 (vs 8192 naive).
// ---------------------------------------------------------------------------
__global__ void combine_kernel(const float* __restrict__ h0,
                               const float* __restrict__ W16,
                               const float* __restrict__ Vseg,
                               float* __restrict__ Hstart) {
  __shared__ float h[64];
  const int tid = threadIdx.x;             // 64
  h[tid] = h0[tid];
  __syncthreads();
  for (int seg = 0; seg < NSEG; ++seg) {
    Hstart[seg * 64 + tid] = h[tid];
    float acc = Vseg[seg * 64 + tid];
    #pragma unroll 8
    for (int k = 0; k < 64; ++k) acc += h[k] * W16[k * 64 + tid];
    __syncthreads();
    h[tid] = acc;
    __syncthreads();
  }
}

// ---------------------------------------------------------------------------
// K6: reconstruct all h_t within segments: h <- h@W + bu_t (16 WMMA steps,
// batched over 16 segment-rows per block), stream results to h_hist.
// ---------------------------------------------------------------------------
__global__ void reconstruct_kernel(const float* __restrict__ Bu,
                                   const float* __restrict__ W,
                                   const float* __restrict__ Hstart,
                                   float* __restrict__ h_hist) {
  __shared__ float Wl[64 * 64];
  __shared__ float Vt[16 * 65];
  const int tid = threadIdx.x;             // 128
  for (int i = tid; i < 64 * 64; i += 128) Wl[i] = W[i];
  const int seg0 = blockIdx.x * 16;
  for (int i = tid; i < 16 * 64; i += 128) {
    const int r = i >> 6, c = i & 63;
    Vt[r * 65 + c] = Hstart[(seg0 + r) * 64 + c];
  }
  __syncthreads();
  const int wave = tid >> 5, lane = tid & 31;
  const int nt = wave, half = lane >> 4, l15 = lane & 15;
  for (int j = 0; j < SEG; ++j) {
    v8f acc;
    #pragma unroll
    for (int r = 0; r < 8; ++r)
      acc[r] = Bu[((seg0 + half * 8 + r) * SEG + j) * 64 + nt * 16 + l15];
    #pragma unroll
    for (int kt = 0; kt < 16; ++kt) {
      v2f a, b;
      a.x = Vt[l15 * 65 + kt * 4 + half * 2 + 0];
      a.y = Vt[l15 * 65 + kt * 4 + half * 2 + 1];
      b.x = Wl[(kt * 4 + half * 2 + 0) * 64 + nt * 16 + l15];
      b.y = Wl[(kt * 4 + half * 2 + 1) * 64 + nt * 16 + l15];
      acc = wmma4(a, b, acc);
    }
    __syncthreads();
    #pragma unroll
    for (int r = 0; r < 8; ++r) {
      const int row = half * 8 + r;
      const int t   = (seg0 + row) * SEG + j;
      Vt[row * 65 + nt * 16 + l15] = acc[r];
      h_hist[(size_t)t * 64 + nt * 16 + l15] = acc[r];
    }
    __syncthreads();
  }
}

// ---------------------------------------------------------------------------
// K7: y = h_hist @ C^T + u @ D^T. WMMA over K=64 with N padded to 16
// (6 valid cols); the tiny u@D^T term added scalar at writeback.
// ---------------------------------------------------------------------------
__global__ void output_kernel(const float* __restrict__ h_hist,
                              const float* __restrict__ Cw,   // [64,16]
                              const float* __restrict__ u_ws, // [T,8]
                              const float* __restrict__ Dm,   // [6,6]
                              float* __restrict__ y_out) {    // [T,6]
  const int t0 = blockIdx.x * 16;
  const int lane = threadIdx.x;            // 32 = 1 wave
  const int half = lane >> 4, l15 = lane & 15;
  v8f acc = {0.f, 0.f, 0.f, 0.f, 0.f, 0.f, 0.f, 0.f};
  #pragma unroll
  for (int kt = 0; kt < 16; ++kt) {
    v2f a, b;
    a.x = h_hist[(size_t)(t0 + l15) * 64 + kt * 4 + half * 2 + 0];
    a.y = h_hist[(size_t)(t0 + l15) * 64 + kt * 4 + half * 2 + 1];
    b.x = Cw[(kt * 4 + half * 2 + 0) * 16 + l15];
    b.y = Cw[(kt * 4 + half * 2 + 1) * 16 + l15];
    acc = wmma4(a, b, acc);
  }
  const int n = l15;
  if (n < P_DIM) {                         // divergence only after WMMA loop
    #pragma unroll
    for (int r = 0; r < 8; ++r) {
      const int t = t0 + half * 8 + r;
      float y = acc[r];
      #pragma unroll
      for (int p = 0; p < P_DIM; ++p) y += u_ws[t * 8 + p] * Dm[n * P_DIM + p];
      y_out[t * P_DIM + n] = y;
    }
  }
}

// ---------------------------------------------------------------------------
extern "C" void kernel_launch(void* const* d_in, const int* in_sizes, int n_in,
                              void* d_out, int out_size, void* d_ws,
                              size_t ws_size, hipStream_t stream) {
  (void)in_sizes; (void)n_in; (void)out_size; (void)ws_size;
  const float* omega = (const float*)d_in[0];
  const float* phi   = (const float*)d_in[1];
  const float* amp   = (const float*)d_in[2];
  const float* h0    = (const float*)d_in[3];
  const float* lam   = (const float*)d_in[4];
  const float* U     = (const float*)d_in[5];
  const float* V     = (const float*)d_in[6];
  const float* Bm    = (const float*)d_in[7];
  const float* Cm    = (const float*)d_in[8];
  const float* Dm    = (const float*)d_in[9];
  const float* dtp   = (const float*)d_in[10];

  float* out    = (float*)d_out;
  float* h_hist = out;                                  // [T,64]
  float* y_out  = out + (size_t)T_STEPS * H_DIM;        // [T,6]
  float* u_out  = y_out + (size_t)T_STEPS * P_DIM;      // [T,6]

  float* f    = (float*)d_ws;
  float* Bu   = f;  f += (size_t)T_STEPS * H_DIM;       // 524288
  float* W    = f;  f += 64 * 64;                       // M^T
  float* Wa   = f;  f += 64 * 64;
  float* Wb   = f;  f += 64 * 64;
  float* Wc   = f;  f += 64 * 64;
  float* Vseg = f;  f += NSEG * 64;
  float* Hst  = f;  f += NSEG * 64;
  float* Cw   = f;  f += 64 * 16;
  float* u_ws = f;  f += (size_t)T_STEPS * 8;

  summarize_kernel<<<T_STEPS, 256, 0, stream>>>(omega, phi, amp, Bm, dtp,
                                                u_out, u_ws, Bu);
  build_kernel<<<20, 256, 0, stream>>>(lam, U, V, Cm, dtp, W, Cw);
  gemm64_kernel<<<1, 512, 0, stream>>>(W,  W,  Wa);     // W^2
  gemm64_kernel<<<1, 512, 0, stream>>>(Wa, Wa, Wb);     // W^4
  gemm64_kernel<<<1, 512, 0, stream>>>(Wb, Wb, Wc);     // W^8
  gemm64_kernel<<<1, 512, 0, stream>>>(Wc, Wc, Wa);     // W^16
  segment_v_kernel<<<SEG_BLOCKS, 128, 0, stream>>>(Bu, W, Vseg);
  combine_kernel<<<1, 64, 0, stream>>>(h0, Wa, Vseg, Hst);
  reconstruct_kernel<<<SEG_BLOCKS, 128, 0, stream>>>(Bu, W, Hst, h_hist);
  output_kernel<<<T_STEPS / 16, 32, 0, stream>>>(h_hist, Cw, u_ws, Dm, y_out);
}